// ProposalTarget_88098369176097
// MI455X (gfx1250) — compile-verified
//
#include <hip/hip_runtime.h>
#include <stdint.h>

#define NBUCKET   1024
#define CAP       4096
#define GT_MAX    128
#define K_PER     128
#define FG_THR    0.5f
#define BG_HI     0.5f
#define BG_LO     0.1f
#define NUM_CLASSES 21

typedef unsigned int v4u __attribute__((ext_vector_type(4)));
typedef int          v8i __attribute__((ext_vector_type(8)));
typedef int          v4i __attribute__((ext_vector_type(4)));

#if __has_builtin(__builtin_amdgcn_tensor_load_to_lds) && __has_builtin(__builtin_amdgcn_s_wait_tensorcnt)
#define USE_TDM 1
#else
#define USE_TDM 0
#endif

// Deterministic per-index uniform in [0,1) (stand-in for the JAX threefry stream).
__device__ __forceinline__ float hash_u(unsigned i) {
    unsigned x = i ^ 0x2545F491u;
    x *= 0x9E3779B9u;
    x ^= x >> 16; x *= 0x7FEB352Du;
    x ^= x >> 15; x *= 0x846CA68Bu;
    x ^= x >> 16;
    return (float)(x >> 8) * (1.0f / 16777216.0f);
}

// ---------------------------------------------------------------- init ------
__global__ void pt_init_kernel(float* out, int out_n,
                               int* hist, int hist_n,
                               int* cnt, int* keep, int* validfg) {
    int i = blockIdx.x * blockDim.x + threadIdx.x;
    if (i < out_n)  out[i]  = 0.0f;
    if (i < hist_n) hist[i] = 0;
    if (i < 2)      cnt[i]  = 0;
    if (i < 2 * K_PER) keep[i] = 0;
    if (i < K_PER)  validfg[i] = 0;
}

// ----------------------------------------------------------- IoU/argmax -----
// Two ROIs per thread; GT tile staged in LDS via the Tensor Data Mover.
// Division-free running argmax: compare inter*bestDenom > bestInter*denom.
__device__ __forceinline__ void load_roi(const float* __restrict__ all_rois,
                                         const float* __restrict__ gt_boxes,
                                         int i, int N,
                                         float& x1, float& y1, float& x2, float& y2) {
    if (i < N) {
        const float* r = all_rois + (size_t)i * 5;
        x1 = r[1]; y1 = r[2]; x2 = r[3]; y2 = r[4];
    } else {
        const float* r = gt_boxes + (size_t)(i - N) * 4;
        x1 = r[0]; y1 = r[1]; x2 = r[2]; y2 = r[3];
    }
}

__global__ void pt_iou_kernel(const float* __restrict__ all_rois,
                              const float* __restrict__ gt_boxes,
                              float* __restrict__ max_ov,
                              int*   __restrict__ assign,
                              int N, int G) {
    __shared__ float4 sgt4[GT_MAX];      // {x1,y1,x2,y2} per GT box
    __shared__ float  sarea[GT_MAX];

#if USE_TDM
    if (threadIdx.x == 0) {
        unsigned lds = (unsigned)(uintptr_t)(void*)sgt4;          // LDS aperture: low 32 bits = LDS addr
        unsigned long long ga = (unsigned long long)(uintptr_t)gt_boxes;
        unsigned nelem = (unsigned)(G * 4);                       // f32 elements in the tile
        v4u g0;
        g0[0] = 1u;                                               // count=1 (valid), user mode
        g0[1] = lds;                                              // lds_addr
        g0[2] = (unsigned)(ga & 0xFFFFFFFFu);                     // global_addr[31:0]
        g0[3] = (unsigned)((ga >> 32) & 0x1FFFFFFu) | (2u << 30); // global_addr[56:32] | type=2
        v8i g1;
        g1[0] = (int)(2u << 16);                                  // data_size=4B, wg_mask=0
        g1[1] = (int)((nelem & 0xFFFFu) << 16);                   // tensor_dim0[15:0]
        g1[2] = (int)((nelem >> 16) | (1u << 16));                // tensor_dim0[31:16] | tensor_dim1=1
        g1[3] = (int)((nelem & 0xFFFFu) << 16);                   // tile_dim0
        g1[4] = 1;                                                // tile_dim1=1, tile_dim2=0
        g1[5] = (int)nelem;                                       // tensor_dim0_stride[31:0]
        g1[6] = 0;
        g1[7] = 0;
        v4i g2 = {0, 0, 0, 0};
        v4i g3 = {0, 0, 0, 0};
        v8i g4 = {0, 0, 0, 0, 0, 0, 0, 0};                        // 6-arg form (clang-23 lane)
        __builtin_amdgcn_tensor_load_to_lds(g0, g1, g2, g3, g4, 0);
        __builtin_amdgcn_s_wait_tensorcnt(0);
    }
#else
    {
        float* s = (float*)sgt4;
        for (int t = threadIdx.x; t < G * 4; t += blockDim.x) s[t] = gt_boxes[t];
    }
#endif
    __syncthreads();

    if (threadIdx.x < (unsigned)G) {
        float4 b = sgt4[threadIdx.x];
        sarea[threadIdx.x] = (b.z - b.x + 1.0f) * (b.w - b.y + 1.0f);
    }
    __syncthreads();

    int total = N + G;
    int S  = gridDim.x * blockDim.x;     // slot stride; each thread owns i0 and i0+S
    int i0 = blockIdx.x * blockDim.x + threadIdx.x;
    int i1 = i0 + S;
    bool a0 = (i0 < total);
    bool a1 = (i1 < total);
    if (!a0) return;

    float p0x1, p0y1, p0x2, p0y2, p1x1 = 0, p1y1 = 0, p1x2 = 0, p1y2 = 0;
    load_roi(all_rois, gt_boxes, i0, N, p0x1, p0y1, p0x2, p0y2);
    if (a1) load_roi(all_rois, gt_boxes, i1, N, p1x1, p1y1, p1x2, p1y2);

    float area0 = (p0x2 - p0x1 + 1.0f) * (p0y2 - p0y1 + 1.0f);
    float area1 = (p1x2 - p1x1 + 1.0f) * (p1y2 - p1y1 + 1.0f);

    float bI0 = -1.0f, bD0 = 1.0f; int bi0 = 0;   // best (inter, denom) for roi0
    float bI1 = -1.0f, bD1 = 1.0f; int bi1 = 0;

#pragma unroll 2
    for (int g = 0; g < G; ++g) {
        float4 b  = sgt4[g];             // ds_load_b128 broadcast (all lanes same addr)
        float  sA = sarea[g];

        float iw0 = fminf(p0x2, b.z) - fmaxf(p0x1, b.x) + 1.0f;
        float ih0 = fminf(p0y2, b.w) - fmaxf(p0y1, b.y) + 1.0f;
        iw0 = fmaxf(iw0, 0.0f); ih0 = fmaxf(ih0, 0.0f);
        float in0 = iw0 * ih0;
        float dn0 = area0 + sA - in0;
        if (in0 * bD0 > bI0 * dn0) { bI0 = in0; bD0 = dn0; bi0 = g; }

        float iw1 = fminf(p1x2, b.z) - fmaxf(p1x1, b.x) + 1.0f;
        float ih1 = fminf(p1y2, b.w) - fmaxf(p1y1, b.y) + 1.0f;
        iw1 = fmaxf(iw1, 0.0f); ih1 = fmaxf(ih1, 0.0f);
        float in1 = iw1 * ih1;
        float dn1 = area1 + sA - in1;
        if (in1 * bD1 > bI1 * dn1) { bI1 = in1; bD1 = dn1; bi1 = g; }
    }

    max_ov[i0] = bI0 / bD0;              // one divide per ROI (was one per pair)
    assign[i0] = bi0;
    if (a1) {
        max_ov[i1] = bI1 / bD1;
        assign[i1] = bi1;
    }
}

// ------------------------------------------------------------- histogram ----
// Pool buckets: bucket 0 = "filler" (-1 key, not in mask); 1..NBUCKET = u-quantile.
__device__ __forceinline__ int ubucket(float u) {
    int b = (int)(u * (float)NBUCKET);
    return 1 + (b < 0 ? 0 : (b > NBUCKET - 1 ? NBUCKET - 1 : b));
}

__global__ void pt_hist_kernel(const float* __restrict__ max_ov, int total,
                               int* __restrict__ hist) {
    __shared__ int lh[2 * (NBUCKET + 1)];
    for (int t = threadIdx.x; t < 2 * (NBUCKET + 1); t += blockDim.x) lh[t] = 0;
    __syncthreads();

    int stride = gridDim.x * blockDim.x;
    for (int i = blockIdx.x * blockDim.x + threadIdx.x; i < total; i += stride) {
        float mv = max_ov[i];
        float u  = hash_u((unsigned)i);
        int bu = ubucket(u);
        bool fg = (mv >= FG_THR);
        bool bg = (mv < BG_HI) && (mv >= BG_LO);
        atomicAdd(&lh[fg ? bu : 0], 1);
        atomicAdd(&lh[(NBUCKET + 1) + (bg ? bu : 0)], 1);
    }
    __syncthreads();
    for (int t = threadIdx.x; t < 2 * (NBUCKET + 1); t += blockDim.x)
        if (lh[t]) atomicAdd(&hist[t], lh[t]);
}

// ------------------------------------------------------------- threshold ----
__global__ void pt_thresh_kernel(const int* __restrict__ hist, int* __restrict__ thr) {
    if (threadIdx.x == 0 && blockIdx.x == 0) {
        for (int c = 0; c < 2; ++c) {
            const int* h = hist + c * (NBUCKET + 1);
            int acc = 0, t = 0;
            for (int b = NBUCKET; b >= 0; --b) {
                acc += h[b];
                if (acc >= K_PER) { t = b; break; }
            }
            thr[c] = t;
        }
    }
}

// --------------------------------------------------------------- collect ----
__global__ void pt_collect_kernel(const float* __restrict__ max_ov, int total,
                                  const int* __restrict__ thr, int* __restrict__ cnt,
                                  float* __restrict__ cand_u, int* __restrict__ cand_idx) {
    int t0 = thr[0], t1 = thr[1];
    int stride = gridDim.x * blockDim.x;
    for (int i = blockIdx.x * blockDim.x + threadIdx.x; i < total; i += stride) {
        float mv = max_ov[i];
        float u  = hash_u((unsigned)i);
        int bu = ubucket(u);
        bool fg = (mv >= FG_THR);
        bool bg = (mv < BG_HI) && (mv >= BG_LO);
        int bfg = fg ? bu : 0;
        int bbg = bg ? bu : 0;
        if (bfg >= t0) {
            int p = atomicAdd(&cnt[0], 1);
            if (p < CAP) { cand_u[p] = fg ? u : -1.0f; cand_idx[p] = i; }
        }
        if (bbg >= t1) {
            int p = atomicAdd(&cnt[1], 1);
            if (p < CAP) { cand_u[CAP + p] = bg ? u : -1.0f; cand_idx[CAP + p] = i; }
        }
    }
}

// ------------------------------------------------------------------ rank ----
// Exact rank within the candidate set (order-independent -> deterministic keep[]).
__global__ void pt_rank_kernel(const float* __restrict__ cand_u,
                               const int*   __restrict__ cand_idx,
                               const int*   __restrict__ cnt,
                               const float* __restrict__ max_ov,
                               int* __restrict__ keep, int* __restrict__ validfg) {
    int cls = blockIdx.x;                // 0 = fg pool, 1 = bg pool
    const float* U = cand_u  + (size_t)cls * CAP;
    const int*   I = cand_idx + (size_t)cls * CAP;
    int C = cnt[cls]; if (C > CAP) C = CAP;
    int Cpad = (C + 255) & ~255;

    __shared__ float su[256];
    __shared__ int   si[256];

    for (int a = threadIdx.x; a < Cpad; a += 256) {
        bool  act = (a < C);
        float ua  = act ? U[a] : -2.0f;
        int   ia  = act ? I[a] : 0x7fffffff;
        int rank = 0;
        for (int base = 0; base < C; base += 256) {
            __syncthreads();
            int j = base + threadIdx.x;
            su[threadIdx.x] = (j < C) ? U[j] : -3.0f;
            si[threadIdx.x] = (j < C) ? I[j] : 0x7fffffff;
            __syncthreads();
            int lim = C - base; if (lim > 256) lim = 256;
            for (int b = 0; b < lim; ++b) {
                float ub = su[b];
                int   ib = si[b];
                rank += (ub > ua) || (ub == ua && ib < ia);   // top_k tie-break: lower idx first
            }
        }
        if (act && rank < K_PER) {
            keep[cls * K_PER + rank] = ia;
            if (cls == 0) validfg[rank] = (max_ov[ia] >= FG_THR) ? 1 : 0;
        }
    }
}

// -------------------------------------------------------------- finalize ----
// out layout (float): rois[256*5] ++ labels[256] ++ bbox_targets[256*84]
__global__ void pt_finalize_kernel(const float* __restrict__ all_rois,
                                   const float* __restrict__ gt_boxes,
                                   const int*   __restrict__ gt_labels,
                                   const int*   __restrict__ assign,
                                   const int*   __restrict__ keep,
                                   const int*   __restrict__ validfg,
                                   int N, int G, float* __restrict__ out) {
    int r = threadIdx.x;
    if (r >= 2 * K_PER) return;
    int idx = keep[r];

    float row[5];
    if (idx < N) {
        const float* p = all_rois + (size_t)idx * 5;
        row[0] = p[0]; row[1] = p[1]; row[2] = p[2]; row[3] = p[3]; row[4] = p[4];
    } else {
        const float* p = gt_boxes + (size_t)(idx - N) * 4;
        row[0] = 0.0f; row[1] = p[0]; row[2] = p[1]; row[3] = p[2]; row[4] = p[3];
    }

    float* out_rois = out;
    float* out_lab  = out + 2 * K_PER * 5;
    float* out_bt   = out + 2 * K_PER * 5 + 2 * K_PER;

    for (int k = 0; k < 5; ++k) out_rois[r * 5 + k] = row[k];

    int g = assign[idx];
    if (g < 0) g = 0; if (g >= G) g = G - 1;

    int lab = 0;
    if (r < K_PER && validfg[r]) lab = gt_labels[g];
    out_lab[r] = (float)lab;

    if (lab > 0) {
        const float* gp = gt_boxes + (size_t)g * 4;
        float ew  = row[3] - row[1] + 1.0f;
        float eh  = row[4] - row[2] + 1.0f;
        float ecx = row[1] + 0.5f * ew;
        float ecy = row[2] + 0.5f * eh;
        float gw  = gp[2] - gp[0] + 1.0f;
        float gh  = gp[3] - gp[1] + 1.0f;
        float gcx = gp[0] + 0.5f * gw;
        float gcy = gp[1] + 0.5f * gh;
        float* d = out_bt + (size_t)r * (4 * NUM_CLASSES) + lab * 4;
        d[0] = (gcx - ecx) / ew;
        d[1] = (gcy - ecy) / eh;
        d[2] = __logf(gw / ew);
        d[3] = __logf(gh / eh);
    }
}

// ------------------------------------------------------------------ host ----
extern "C" void kernel_launch(void* const* d_in, const int* in_sizes, int n_in,
                              void* d_out, int out_size, void* d_ws, size_t ws_size,
                              hipStream_t stream) {
    const float* all_rois  = (const float*)d_in[0];
    const float* gt_boxes  = (const float*)d_in[1];
    const int*   gt_labels = (const int*)d_in[2];
    int N = in_sizes[0] / 5;
    int G = in_sizes[1] / 4;
    if (G > GT_MAX) G = GT_MAX;
    int total = N + G;
    (void)n_in; (void)ws_size;

    char* ws = (char*)d_ws;
    size_t off = 0;
    float* max_ov  = (float*)(ws + off); off += (size_t)total * 4;
    int*   assign  = (int*)  (ws + off); off += (size_t)total * 4;
    int*   hist    = (int*)  (ws + off); off += (size_t)2 * (NBUCKET + 1) * 4;
    int*   thr     = (int*)  (ws + off); off += 2 * 4;
    int*   cnt     = (int*)  (ws + off); off += 2 * 4;
    int*   keep    = (int*)  (ws + off); off += 2 * K_PER * 4;
    int*   validfg = (int*)  (ws + off); off += K_PER * 4;
    float* cand_u  = (float*)(ws + off); off += (size_t)2 * CAP * 4;
    int*   cand_idx= (int*)  (ws + off); off += (size_t)2 * CAP * 4;

    int hist_n = 2 * (NBUCKET + 1);
    int init_n = out_size > hist_n ? out_size : hist_n;
    int init_blocks = (init_n + 255) / 256;
    pt_init_kernel<<<init_blocks, 256, 0, stream>>>((float*)d_out, out_size,
                                                    hist, hist_n, cnt, keep, validfg);

    int iou_blocks = (total + 511) / 512;        // 2 ROIs per thread
    pt_iou_kernel<<<iou_blocks, 256, 0, stream>>>(all_rois, gt_boxes, max_ov, assign, N, G);

    pt_hist_kernel<<<512, 256, 0, stream>>>(max_ov, total, hist);
    pt_thresh_kernel<<<1, 32, 0, stream>>>(hist, thr);
    pt_collect_kernel<<<512, 256, 0, stream>>>(max_ov, total, thr, cnt, cand_u, cand_idx);
    pt_rank_kernel<<<2, 256, 0, stream>>>(cand_u, cand_idx, cnt, max_ov, keep, validfg);
    pt_finalize_kernel<<<1, 256, 0, stream>>>(all_rois, gt_boxes, gt_labels,
                                              assign, keep, validfg, N, G, (float*)d_out);
}